// LGIFBolck_53068615910118
// MI455X (gfx1250) — compile-verified
//
#include <hip/hip_runtime.h>
#include <math.h>

typedef __attribute__((ext_vector_type(2))) float v2f;
typedef __attribute__((ext_vector_type(8))) float v8f;

#define LN_EPS 1e-5f

// ---------------------------------------------------------------------------
// Block-wide mean/var over 128 channels (blockDim.x == 128 -> 4 wave32s)
// ---------------------------------------------------------------------------
__device__ inline void block_meanvar(float v, float& mu, float& var) {
  __shared__ float s_sum[4], s_sq[4];
  float a = v, b = v * v;
#pragma unroll
  for (int off = 16; off > 0; off >>= 1) {
    a += __shfl_xor(a, off, 32);
    b += __shfl_xor(b, off, 32);
  }
  int wid = threadIdx.x >> 5;
  if ((threadIdx.x & 31) == 0) { s_sum[wid] = a; s_sq[wid] = b; }
  __syncthreads();
  float ts = s_sum[0] + s_sum[1] + s_sum[2] + s_sum[3];
  float tq = s_sq[0]  + s_sq[1]  + s_sq[2]  + s_sq[3];
  __syncthreads();
  mu  = ts * (1.0f / 128.0f);
  var = tq * (1.0f / 128.0f) - mu * mu;
}

// ---------------------------------------------------------------------------
// NT adjacent 16x16 f32 tiles per wave via V_WMMA_F32_16X16X4_F32.
// D[m][n] = sum_k A[m0+m][k] * W[n][k]  for n in [n0, n0+16*NT)
// A lanes: 0-15 -> K=kk+0/+1 (a.x/a.y), 16-31 -> K=kk+2/+3. B symmetric.
// GUARD: W rows >= nValid are clamped to row 0 and multiplied by 0
// (branchless; no exec-mask manipulation in the inner loop).
// ---------------------------------------------------------------------------
template <int NT, bool GUARD>
__device__ inline void wmma_tiles_f32(const float* __restrict__ A, int lda,
                                      const float* __restrict__ W, int ldw,
                                      int m0, int n0, int K, int nValid,
                                      v8f (&acc)[NT]) {
  int lane = threadIdx.x & 31;
  int ml = lane & 15;
  int ks = (lane >> 4) << 1;                 // 0 or 2
  const float* arow = A + (size_t)(m0 + ml) * lda;
  const float* brow[NT];
  float bscale[NT];
#pragma unroll
  for (int j = 0; j < NT; ++j) {
    int nb = n0 + j * 16 + ml;
    bool ok = !GUARD || (nb < nValid);
    brow[j] = W + (size_t)(ok ? nb : 0) * ldw;
    bscale[j] = ok ? 1.0f : 0.0f;
    acc[j] = {};
  }
  for (int kk = 0; kk < K; kk += 4) {
    v2f a;
    a.x = arow[kk + ks];
    a.y = arow[kk + ks + 1];
#pragma unroll
    for (int j = 0; j < NT; ++j) {
      v2f b;
      b.x = brow[j][kk + ks];
      b.y = brow[j][kk + ks + 1];
      if (GUARD) { b.x *= bscale[j]; b.y *= bscale[j]; }   // v_cndmask-free zeroing
      acc[j] = __builtin_amdgcn_wmma_f32_16x16x4_f32(false, a, false, b,
                                                     (short)0, acc[j],
                                                     false, false);
    }
  }
}

// ---------------------------------------------------------------------------
// K1: LN over channels of NCHW input -> NHWC 'inp' buffer
// ---------------------------------------------------------------------------
__global__ void k_ln1(const float* __restrict__ x, const float* __restrict__ w,
                      const float* __restrict__ b, float* __restrict__ inp) {
  int pixel = blockIdx.x;                 // 0..16383  (b*4096 + l)
  int bb = pixel >> 12, l = pixel & 4095;
  int ch = threadIdx.x;
  float v = x[((size_t)(bb * 128 + ch) << 12) + l];
  float mu, var; block_meanvar(v, mu, var);
  inp[(size_t)pixel * 128 + ch] = (v - mu) * rsqrtf(var + LN_EPS) * w[ch] + b[ch];
}

// ---------------------------------------------------------------------------
// K2: depthwise 3x3 (SAME, zero pad) + bias -> LN -> exact GELU
// ---------------------------------------------------------------------------
__global__ void k_dwconv_ln_gelu(const float* __restrict__ inp,
                                 const float* __restrict__ cw,   // (C,1,3,3)
                                 const float* __restrict__ cb,
                                 const float* __restrict__ lnw,
                                 const float* __restrict__ lnb,
                                 float* __restrict__ hout) {
  int pixel = blockIdx.x;
  int bb = pixel >> 12, l = pixel & 4095;
  int hh = l >> 6, ww = l & 63;
  int ch = threadIdx.x;
  float s = cb[ch];
#pragma unroll
  for (int dy = -1; dy <= 1; ++dy) {
#pragma unroll
    for (int dx = -1; dx <= 1; ++dx) {
      int y2 = hh + dy, x2 = ww + dx;
      if (y2 >= 0 && y2 < 64 && x2 >= 0 && x2 < 64)
        s += cw[ch * 9 + (dy + 1) * 3 + (dx + 1)] *
             inp[(size_t)(((bb << 12) + (y2 << 6) + x2)) * 128 + ch];
    }
  }
  float mu, var; block_meanvar(s, mu, var);
  float t = (s - mu) * rsqrtf(var + LN_EPS) * lnw[ch] + lnb[ch];
  hout[(size_t)pixel * 128 + ch] = 0.5f * t * (1.0f + erff(t * 0.70710678118f));
}

// ---------------------------------------------------------------------------
// Generic GEMM: D = A @ W^T (+bias) (+res). Each wave computes NT n-tiles.
// NtGroups = number of NT-wide column groups (N_padded / (16*NT)).
// ---------------------------------------------------------------------------
template <int NT, bool GUARD>
__global__ void k_gemm(const float* __restrict__ A, int lda,
                       const float* __restrict__ W, int ldw,
                       const float* __restrict__ bias,
                       const float* __restrict__ res,
                       float* __restrict__ D, int ldd,
                       int Mtiles, int NtGroups, int K, int nValid) {
  int wid = blockIdx.x * (blockDim.x >> 5) + (threadIdx.x >> 5);
  int tm = wid / NtGroups, tg = wid - tm * NtGroups;
  if (tm >= Mtiles) return;                       // wave-uniform
  int m0 = tm << 4, n0 = tg * (16 * NT);
  v8f acc[NT];
  wmma_tiles_f32<NT, GUARD>(A, lda, W, ldw, m0, n0, K, nValid, acc);
  int lane = threadIdx.x & 31;
  int mb = m0 + ((lane >> 4) << 3);
#pragma unroll
  for (int j = 0; j < NT; ++j) {
    int n = n0 + j * 16 + (lane & 15);
    if (!GUARD || n < nValid) {
      float bv = bias ? bias[n] : 0.0f;
#pragma unroll
      for (int r = 0; r < 8; ++r) {
        size_t idx = (size_t)(mb + r) * ldd + n;
        float v = acc[j][r] + bv;
        if (res) v += res[idx];
        D[idx] = v;
      }
    }
  }
}

// ---------------------------------------------------------------------------
// K4: LN -> write directly into chunked mamba input u[(c*4+b)*4096+l][d]
// ---------------------------------------------------------------------------
__global__ void k_ln2_chunks(const float* __restrict__ y, const float* __restrict__ w,
                             const float* __restrict__ b, float* __restrict__ u) {
  int pixel = blockIdx.x;
  int bb = pixel >> 12, l = pixel & 4095;
  int ch = threadIdx.x;
  float v = y[(size_t)pixel * 128 + ch];
  float mu, var; block_meanvar(v, mu, var);
  float xn = (v - mu) * rsqrtf(var + LN_EPS) * w[ch] + b[ch];
  int c = ch >> 5, d = ch & 31;
  u[(size_t)((((c << 2) + bb) << 12) + l) * 32 + d] = xn;
}

// ---------------------------------------------------------------------------
// K6: causal depthwise conv1d (k=4) + bias + SiLU on first 64 cols of xz
// ---------------------------------------------------------------------------
__global__ void k_conv1d_silu(const float* __restrict__ xz,
                              const float* __restrict__ cw,   // (64,4)
                              const float* __restrict__ cb,
                              float* __restrict__ xin) {
  int gid = blockIdx.x * blockDim.x + threadIdx.x;   // 65536*64 threads
  int row = gid >> 6, d = gid & 63;
  int bc = row >> 12, l = row & 4095;
  float s = cb[d];
#pragma unroll
  for (int j = 0; j < 4; ++j) {
    int lt = l - 3 + j;
    if (lt >= 0)
      s += cw[d * 4 + j] * xz[(size_t)((bc << 12) + lt) * 128 + d];
  }
  xin[(size_t)row * 64 + d] = s / (1.0f + __expf(-s));
}

// ---------------------------------------------------------------------------
// K8: selective scan. One thread per (bc, d); h[16] kept in registers.
// Fuses dt_proj + softplus, state recurrence, C reduction, D-skip, silu(z) gate.
// ---------------------------------------------------------------------------
__global__ void k_scan(const float* __restrict__ xdbl,  // (65536,34): dt|B|C
                       const float* __restrict__ xin,   // (65536,64)
                       const float* __restrict__ xz,    // z at col 64+d
                       const float* __restrict__ dtw,   // (64,2)
                       const float* __restrict__ dtb,   // (64,)
                       const float* __restrict__ A_log, // (64,16)
                       const float* __restrict__ Dskip, // (64,)
                       float* __restrict__ g) {         // (65536,64)
  int bc = blockIdx.x;          // 0..15
  int d  = threadIdx.x;         // 0..63
  float Arow[16];
#pragma unroll
  for (int n = 0; n < 16; ++n) Arow[n] = -__expf(A_log[d * 16 + n]);
  float w0 = dtw[d * 2 + 0], w1 = dtw[d * 2 + 1];
  float b0 = dtb[d], dsk = Dskip[d];
  float h[16];
#pragma unroll
  for (int n = 0; n < 16; ++n) h[n] = 0.0f;
  size_t rbase = (size_t)bc << 12;
  for (int t = 0; t < 4096; ++t) {
    size_t row = rbase + t;
    const float* xd = xdbl + row * 34;
    float dt = w0 * xd[0] + w1 * xd[1] + b0;
    float delta = (dt > 20.0f) ? dt : log1pf(__expf(dt));
    float xv = xin[row * 64 + d];
    float dx = delta * xv;
    float y = 0.0f;
#pragma unroll
    for (int n = 0; n < 16; ++n) {
      float dA = __expf(delta * Arow[n]);
      h[n] = dA * h[n] + dx * xd[2 + n];
      y += h[n] * xd[18 + n];
    }
    y += dsk * xv;
    float z = xz[row * 128 + 64 + d];
    g[row * 64 + d] = y * (z / (1.0f + __expf(-z)));
  }
}

// ---------------------------------------------------------------------------
// K9: out_proj GEMM (NT=2 covers full N=32) fused with skip add and
// chunk->channel scatter into 'cat'
// ---------------------------------------------------------------------------
__global__ void k_gemm_outproj(const float* __restrict__ G,     // (65536,64)
                               const float* __restrict__ W,     // (32,64)
                               const float* __restrict__ U,     // (65536,32)
                               const float* __restrict__ skip,  // [1]
                               float* __restrict__ cat) {       // NHWC (16384,128)
  int tm = blockIdx.x * (blockDim.x >> 5) + (threadIdx.x >> 5);
  if (tm >= 4096) return;
  int m0 = tm << 4;
  v8f acc[2];
  wmma_tiles_f32<2, false>(G, 64, W, 64, m0, 0, 64, 32, acc);
  int lane = threadIdx.x & 31;
  int mb = m0 + ((lane >> 4) << 3);
  float ss = skip[0];
#pragma unroll
  for (int j = 0; j < 2; ++j) {
    int n = j * 16 + (lane & 15);
#pragma unroll
    for (int r = 0; r < 8; ++r) {
      int row = mb + r;                             // 0..65535
      int bc = row >> 12, l = row & 4095;
      int b = bc & 3, c = bc >> 2;
      float v = acc[j][r] + ss * U[(size_t)row * 32 + n];
      cat[(size_t)(((b << 12) + l)) * 128 + (c << 5) + n] = v;
    }
  }
}

// ---------------------------------------------------------------------------
// K10: final LN over (inp + cat)
// ---------------------------------------------------------------------------
__global__ void k_ln_final(const float* __restrict__ inp, const float* __restrict__ cat,
                           const float* __restrict__ w, const float* __restrict__ b,
                           float* __restrict__ t) {
  int pixel = blockIdx.x;
  int ch = threadIdx.x;
  size_t i = (size_t)pixel * 128 + ch;
  float v = inp[i] + cat[i];
  float mu, var; block_meanvar(v, mu, var);
  t[i] = (v - mu) * rsqrtf(var + LN_EPS) * w[ch] + b[ch];
}

// ---------------------------------------------------------------------------
// K11: final projection GEMM (NT=4, 2 column groups) + bias,
// fused NHWC -> NCHW transposed store
// ---------------------------------------------------------------------------
__global__ void k_gemm_final(const float* __restrict__ T,     // (16384,128)
                             const float* __restrict__ W,     // (128,128)
                             const float* __restrict__ bias,
                             float* __restrict__ out) {       // (B,C,H,W)
  int wid = blockIdx.x * (blockDim.x >> 5) + (threadIdx.x >> 5);
  int tm = wid >> 1, tg = wid & 1;                  // 2 groups of 64 cols
  if (tm >= 1024) return;
  int m0 = tm << 4, n0 = tg << 6;
  v8f acc[4];
  wmma_tiles_f32<4, false>(T, 128, W, 128, m0, n0, 128, 128, acc);
  int lane = threadIdx.x & 31;
  int mb = m0 + ((lane >> 4) << 3);
#pragma unroll
  for (int j = 0; j < 4; ++j) {
    int n = n0 + j * 16 + (lane & 15);
    float bv = bias[n];
#pragma unroll
    for (int r = 0; r < 8; ++r) {
      int row = mb + r;                             // 0..16383
      int b = row >> 12, l = row & 4095;
      out[((size_t)(b * 128 + n) << 12) + l] = acc[j][r] + bv;
    }
  }
}

// ---------------------------------------------------------------------------
extern "C" void kernel_launch(void* const* d_in, const int* in_sizes, int n_in,
                              void* d_out, int out_size, void* d_ws, size_t ws_size,
                              hipStream_t stream) {
  (void)in_sizes; (void)n_in; (void)out_size; (void)ws_size;
  const float* x         = (const float*)d_in[0];
  const float* ln_w      = (const float*)d_in[1];
  const float* ln_b      = (const float*)d_in[2];
  const float* lc_fc1_w  = (const float*)d_in[3];
  const float* lc_fc1_b  = (const float*)d_in[4];
  const float* lc_ln_w   = (const float*)d_in[5];
  const float* lc_ln_b   = (const float*)d_in[6];
  const float* lc_fc2_w  = (const float*)d_in[7];
  const float* lc_fc2_b  = (const float*)d_in[8];
  const float* in_proj_w = (const float*)d_in[9];
  const float* conv1d_w  = (const float*)d_in[10];
  const float* conv1d_b  = (const float*)d_in[11];
  const float* x_proj_w  = (const float*)d_in[12];
  const float* dt_proj_w = (const float*)d_in[13];
  const float* dt_proj_b = (const float*)d_in[14];
  const float* A_log     = (const float*)d_in[15];
  const float* D_skip    = (const float*)d_in[16];
  const float* out_proj_w= (const float*)d_in[17];
  const float* proj_w    = (const float*)d_in[18];
  const float* proj_b    = (const float*)d_in[19];
  const float* skip_sc   = (const float*)d_in[20];
  float* out = (float*)d_out;

  // workspace layout (floats)
  float* ws   = (float*)d_ws;
  float* inp  = ws;                   // 2,097,152  LN1 output (NHWC)
  float* ybuf = inp  + 2097152;       // 2,097,152  LC-block output
  float* hbuf = ybuf + 2097152;       // 2,097,152  dwconv/GELU; reused as final-LN out
  float* ubuf = hbuf + 2097152;       // 2,097,152  mamba chunks u (65536,32)
  float* xz   = ubuf + 2097152;       // 8,388,608  in_proj out (65536,128)
  float* xin  = xz   + 8388608;       // 4,194,304  conv1d+silu (65536,64)
  float* xdbl = xin  + 4194304;       // 2,228,224  x_proj out (65536,34)
  float* gbuf = xdbl + 2228224;       // 4,194,304  gated scan out (65536,64)
  float* cat  = gbuf + 4194304;       // 2,097,152  mamba concat (NHWC)

  dim3 blk128(128);
  // 1) LN1
  k_ln1<<<16384, blk128, 0, stream>>>(x, ln_w, ln_b, inp);
  // 2) depthwise 3x3 + LN + GELU
  k_dwconv_ln_gelu<<<16384, blk128, 0, stream>>>(inp, lc_fc1_w, lc_fc1_b,
                                                 lc_ln_w, lc_ln_b, hbuf);
  // 3) y = h @ fc2^T + b + inp   (NT=4: 1024 row-tiles x 2 col groups)
  k_gemm<4, false><<<(1024 * 2 + 7) / 8, 256, 0, stream>>>(
      hbuf, 128, lc_fc2_w, 128, lc_fc2_b, inp, ybuf, 128, 1024, 2, 128, 128);
  // 4) LN2 + chunk gather -> u
  k_ln2_chunks<<<16384, blk128, 0, stream>>>(ybuf, ln_w, ln_b, ubuf);
  // 5) xz = u @ in_proj^T   (NT=4: 4096 row-tiles x 2 col groups)
  k_gemm<4, false><<<(4096 * 2 + 7) / 8, 256, 0, stream>>>(
      ubuf, 32, in_proj_w, 32, nullptr, nullptr, xz, 128, 4096, 2, 32, 128);
  // 6) causal conv1d + SiLU
  k_conv1d_silu<<<(65536 * 64) / 256, 256, 0, stream>>>(xz, conv1d_w, conv1d_b, xin);
  // 7) x_dbl = xin @ x_proj^T  (NT=3 guarded: N padded 48, valid 34)
  k_gemm<3, true><<<(4096 * 1 + 7) / 8, 256, 0, stream>>>(
      xin, 64, x_proj_w, 64, nullptr, nullptr, xdbl, 34, 4096, 1, 64, 34);
  // 8) selective scan + gate
  k_scan<<<16, 64, 0, stream>>>(xdbl, xin, xz, dt_proj_w, dt_proj_b,
                                A_log, D_skip, gbuf);
  // 9) out_proj + skip + scatter into cat (NT=2 covers N=32)
  k_gemm_outproj<<<(4096 + 7) / 8, 256, 0, stream>>>(gbuf, out_proj_w,
                                                     ubuf, skip_sc, cat);
  // 10) final LN over inp + cat
  k_ln_final<<<16384, blk128, 0, stream>>>(inp, cat, ln_w, ln_b, hbuf);
  // 11) final projection + transpose to NCHW (NT=4, 2 col groups)
  k_gemm_final<<<(1024 * 2 + 7) / 8, 256, 0, stream>>>(hbuf, proj_w, proj_b, out);
}